// LoRATinyGNN_38285338476795
// MI455X (gfx1250) — compile-verified
//
#include <hip/hip_runtime.h>

#define N_NODES 100000
#define N_EDGES 1600000
#define HID 128
#define FIN 15
#define FINP 16
#define LORA_SCALE 0.25f
#define LN_EPS 1e-5f

typedef __attribute__((ext_vector_type(2))) float v2f;
typedef __attribute__((ext_vector_type(8))) float v8f;

// ---------------- degree / dinv ----------------
__global__ __launch_bounds__(256) void k_zero(float* __restrict__ p, int n) {
  int i = blockIdx.x * 256 + threadIdx.x;
  if (i < n) p[i] = 0.f;
}

__global__ __launch_bounds__(256) void k_deg(const int* __restrict__ ei, float* __restrict__ deg) {
  int e = blockIdx.x * 256 + threadIdx.x;
  if (e < N_EDGES) atomicAdd(&deg[ei[N_EDGES + e]], 1.0f);
}

__global__ __launch_bounds__(256) void k_dinv(float* __restrict__ d) {
  int i = blockIdx.x * 256 + threadIdx.x;
  if (i < N_NODES) d[i] = rsqrtf(d[i] + 1.0f);
}

// ---------- effective LoRA weight: We(Dout x Dpad) = Wb + scale*A@B^T, zero pad ----------
__global__ __launch_bounds__(256) void k_eff_w(const float* __restrict__ Wb, const float* __restrict__ A,
                                               const float* __restrict__ B, float* __restrict__ We,
                                               int Dout, int Din, int Dpad) {
  int i = blockIdx.x * 256 + threadIdx.x;
  if (i >= Dout * Dpad) return;
  int o = i / Dpad, c = i % Dpad;
  float v = 0.f;
  if (c < Din) {
    float l = A[o*4+0]*B[c*4+0] + A[o*4+1]*B[c*4+1] + A[o*4+2]*B[c*4+2] + A[o*4+3]*B[c*4+3];
    v = Wb[o*Din + c] + LORA_SCALE * l;
  }
  We[i] = v;
}

// ---------------- aggregation, F=15 (padded to 16) ----------------
__global__ __launch_bounds__(256) void k_self15(const float* __restrict__ nf, const float* __restrict__ dinv,
                                                float* __restrict__ out) {
  int i = blockIdx.x * 256 + threadIdx.x;
  if (i >= N_NODES * FINP) return;
  int n = i >> 4, f = i & 15;
  float d = dinv[n];
  out[i] = (f < FIN) ? nf[(size_t)n * FIN + f] * d * d : 0.f;
}

__global__ __launch_bounds__(256) void k_scat15(const int* __restrict__ ei, const float* __restrict__ nf,
                                                const float* __restrict__ dinv, float* __restrict__ out) {
  int i = blockIdx.x * 256 + threadIdx.x;
  if (i >= N_EDGES * FINP) return;
  int e = i >> 4, f = i & 15;
  if (f >= FIN) return;
  int s = ei[e], d = ei[N_EDGES + e];
  float w = dinv[s] * dinv[d];
  atomicAdd(&out[(size_t)d * FINP + f], nf[(size_t)s * FIN + f] * w);
}

// ---------------- aggregation, F=128 ----------------
__global__ __launch_bounds__(256) void k_self128(const float* __restrict__ h, const float* __restrict__ dinv,
                                                 float* __restrict__ out) {
  size_t i = (size_t)blockIdx.x * 256 + threadIdx.x;
  if (i >= (size_t)N_NODES * HID) return;
  int n = (int)(i >> 7);
  float d = dinv[n];
  out[i] = h[i] * d * d;
}

// one wave per edge: lane carries float4 (512B coalesced row), 4 fp32 atomics into dst row
__global__ __launch_bounds__(256) void k_scat128(const int* __restrict__ ei, const float* __restrict__ h,
                                                 const float* __restrict__ dinv, float* __restrict__ out) {
  int e = blockIdx.x * 8 + (threadIdx.x >> 5);
  if (e >= N_EDGES) return;
  int lane = threadIdx.x & 31;
  int s = ei[e], d = ei[N_EDGES + e];
  float w = dinv[s] * dinv[d];
  const float4* hs = (const float4*)(h + (size_t)s * HID);
  float4 v = hs[lane];
  float* o = out + (size_t)d * HID + lane * 4;
  atomicAdd(o + 0, v.x * w);
  atomicAdd(o + 1, v.y * w);
  atomicAdd(o + 2, v.z * w);
  atomicAdd(o + 3, v.w * w);
}

// ---------------- fused WMMA GEMM (Y = X @ W^T) + LayerNorm + ReLU ----------------
// X: N x K (K = 16 or 128), W: 128 x K, Y: N x 128. Block = 256 threads = 8 waves.
// The 16 x K A-tile of X is shared by all 8 waves -> stage it once in LDS using the
// CDNA5 async global->LDS path (ASYNCcnt), then feed A fragments from LDS.
__global__ __launch_bounds__(256) void k_gemm_ln_relu(const float* __restrict__ X, const float* __restrict__ W,
                                                      const float* __restrict__ g, const float* __restrict__ b,
                                                      float* __restrict__ Y, int K) {
  __shared__ float xs[16 * 128];    // staged X tile (16 x K), contiguous in global
  __shared__ float tile[16 * 128];  // output tile before LN
  __shared__ float stats[32];       // mean, rstd per row

  const int row0 = blockIdx.x * 16;

  // ---- async-stage the contiguous 16*K-float X tile into LDS ----
  {
    const float* src = X + (size_t)row0 * K;   // 16 rows of K floats, contiguous
    const int nvec = 4 * K;                    // number of 16-byte chunks (4K float4s)
    for (int t = threadIdx.x; t < nvec; t += 256) {
      unsigned long long ga = (unsigned long long)(uintptr_t)(src + t * 4);
      unsigned int la = (unsigned int)(uintptr_t)(&xs[t * 4]);  // low 32 bits = LDS byte addr
      asm volatile("global_load_async_to_lds_b128 %0, %1, off"
                   :: "v"(la), "v"(ga) : "memory");
    }
    asm volatile("s_wait_asynccnt 0" ::: "memory");
  }
  __syncthreads();

  const int wave    = threadIdx.x >> 5;
  const int lane    = threadIdx.x & 31;
  const int colbase = wave * 16;
  const int mrow    = lane & 15;          // M (for A) / N (for B) within tile
  const int khalf   = (lane >> 4) * 2;    // lanes 0-15: K{0,1}; lanes 16-31: K{2,3}

  const float* xrow = xs + mrow * K;                                // A row (LDS)
  const float* __restrict__ wrow = W + (size_t)(colbase + mrow) * K;  // B col n = W row n

  v8f c = {};
  for (int k = 0; k < K; k += 4) {
    v2f a, bf;
    a.x  = xrow[k + khalf];  a.y  = xrow[k + khalf + 1];
    bf.x = wrow[k + khalf];  bf.y = wrow[k + khalf + 1];
    c = __builtin_amdgcn_wmma_f32_16x16x4_f32(false, a, false, bf, (short)0, c, false, false);
  }

  // D layout: VGPR i -> lanes 0-15: D[M=i][N=lane]; lanes 16-31: D[M=i+8][N=lane-16]
  const int n     = lane & 15;
  const int mbase = (lane < 16) ? 0 : 8;
#pragma unroll
  for (int i = 0; i < 8; ++i)
    tile[(mbase + i) * 128 + colbase + n] = c[i];
  __syncthreads();

  if (threadIdx.x < 16) {
    int r = threadIdx.x;
    float s = 0.f, s2 = 0.f;
    const float4* t = (const float4*)(tile + r * 128);
#pragma unroll 8
    for (int j = 0; j < 32; ++j) {
      float4 v = t[j];
      s  += v.x + v.y + v.z + v.w;
      s2 += v.x * v.x + v.y * v.y + v.z * v.z + v.w * v.w;
    }
    float mean = s * (1.f / 128.f);
    float var  = s2 * (1.f / 128.f) - mean * mean;
    stats[r * 2]     = mean;
    stats[r * 2 + 1] = rsqrtf(var + LN_EPS);
  }
  __syncthreads();

#pragma unroll
  for (int i = 0; i < 8; ++i) {
    int flat = threadIdx.x + i * 256;
    int r    = flat >> 7;
    int col  = flat & 127;
    float x  = tile[flat];
    float y  = (x - stats[r * 2]) * stats[r * 2 + 1] * g[col] + b[col];
    Y[(size_t)row0 * 128 + flat] = fmaxf(y, 0.f);
  }
}

// ---------------- edge classifier: logits[e][c] = sum_j h[s][j]*h[d][j]*Wc[c][j] ----------------
__global__ __launch_bounds__(256) void k_edge_cls(const int* __restrict__ ei, const float* __restrict__ h,
                                                  const float* __restrict__ Wc, float* __restrict__ logits) {
  int e = blockIdx.x * 8 + (threadIdx.x >> 5);
  if (e >= N_EDGES) return;
  int lane = threadIdx.x & 31;
  const float4* hs = (const float4*)(h + (size_t)ei[e] * HID);
  const float4* hd = (const float4*)(h + (size_t)ei[N_EDGES + e] * HID);
  const float4* w0 = (const float4*)(Wc);
  const float4* w1 = (const float4*)(Wc + HID);
  float4 a = hs[lane], bb = hd[lane], c0 = w0[lane], c1 = w1[lane];
  float px = a.x * bb.x, py = a.y * bb.y, pz = a.z * bb.z, pw = a.w * bb.w;
  float d0 = px * c0.x + py * c0.y + pz * c0.z + pw * c0.w;
  float d1 = px * c1.x + py * c1.y + pz * c1.z + pw * c1.w;
#pragma unroll
  for (int off = 16; off > 0; off >>= 1) {
    d0 += __shfl_down(d0, off, 32);
    d1 += __shfl_down(d1, off, 32);
  }
  if (lane == 0) {
    logits[(size_t)e * 2 + 0] = d0;
    logits[(size_t)e * 2 + 1] = d1;
  }
}

extern "C" void kernel_launch(void* const* d_in, const int* in_sizes, int n_in,
                              void* d_out, int out_size, void* d_ws, size_t ws_size,
                              hipStream_t stream) {
  const float* node_feat = (const float*)d_in[0];
  const int*   ei        = (const int*)d_in[1];
  const float* W1b = (const float*)d_in[2];
  const float* W2b = (const float*)d_in[3];
  const float* W3b = (const float*)d_in[4];
  const float* Wcb = (const float*)d_in[5];
  const float* A1 = (const float*)d_in[6],  *B1 = (const float*)d_in[7];
  const float* A2 = (const float*)d_in[8],  *B2 = (const float*)d_in[9];
  const float* A3 = (const float*)d_in[10], *B3 = (const float*)d_in[11];
  const float* Ac = (const float*)d_in[12], *Bc = (const float*)d_in[13];
  const float* g1 = (const float*)d_in[14], *b1 = (const float*)d_in[15];
  const float* g2 = (const float*)d_in[16], *b2 = (const float*)d_in[17];
  const float* g3 = (const float*)d_in[18], *b3 = (const float*)d_in[19];

  float* out_f  = (float*)d_out;
  float* logits = out_f;                          // E*2
  float* h_out  = out_f + (size_t)N_EDGES * 2;    // N*128 (layer-3 output lives here)

  // workspace layout (floats)
  float* ws   = (float*)d_ws;
  float* dinv = ws;                               // N (padded to 102400)
  float* bufA = ws + 102400;                      // N*128 (also aliased as N*16 agg15)
  float* bufB = bufA + (size_t)N_NODES * HID;     // N*128
  float* W1e  = bufB + (size_t)N_NODES * HID;     // 128*16
  float* W2e  = W1e + 128 * FINP;                 // 128*128
  float* W3e  = W2e + 128 * 128;                  // 128*128
  float* Wce  = W3e + 128 * 128;                  // 2*128
  float* agg15 = bufA;                            // N*16 view

  const int TB = 256;
  const int blkN      = (N_NODES + TB - 1) / TB;
  const int blkE      = (N_EDGES + TB - 1) / TB;
  const int blkN16    = (N_NODES * FINP + TB - 1) / TB;
  const int blkE16    = (N_EDGES * FINP + TB - 1) / TB;
  const int blkNH     = (int)(((size_t)N_NODES * HID + TB - 1) / TB);
  const int blkWaveE  = (N_EDGES + 7) / 8;
  const int blkGemm   = N_NODES / 16;  // 6250 exactly

  // degree -> dinv
  k_zero<<<blkN, TB, 0, stream>>>(dinv, N_NODES);
  k_deg<<<blkE, TB, 0, stream>>>(ei, dinv);
  k_dinv<<<blkN, TB, 0, stream>>>(dinv);

  // effective LoRA weights
  k_eff_w<<<(128 * FINP + TB - 1) / TB, TB, 0, stream>>>(W1b, A1, B1, W1e, 128, FIN, FINP);
  k_eff_w<<<(128 * 128 + TB - 1) / TB, TB, 0, stream>>>(W2b, A2, B2, W2e, 128, 128, 128);
  k_eff_w<<<(128 * 128 + TB - 1) / TB, TB, 0, stream>>>(W3b, A3, B3, W3e, 128, 128, 128);
  k_eff_w<<<(2 * 128 + TB - 1) / TB, TB, 0, stream>>>(Wcb, Ac, Bc, Wce, 2, 128, 128);

  // layer 1: agg(node_feat) -> agg15; gemm(K=16)+LN+ReLU -> bufB
  k_self15<<<blkN16, TB, 0, stream>>>(node_feat, dinv, agg15);
  k_scat15<<<blkE16, TB, 0, stream>>>(ei, node_feat, dinv, agg15);
  k_gemm_ln_relu<<<blkGemm, TB, 0, stream>>>(agg15, W1e, g1, b1, bufB, FINP);

  // layer 2: agg(bufB) -> bufA; gemm(K=128)+LN+ReLU -> bufB
  k_self128<<<blkNH, TB, 0, stream>>>(bufB, dinv, bufA);
  k_scat128<<<blkWaveE, TB, 0, stream>>>(ei, bufB, dinv, bufA);
  k_gemm_ln_relu<<<blkGemm, TB, 0, stream>>>(bufA, W2e, g2, b2, bufB, HID);

  // layer 3: agg(bufB) -> bufA; gemm -> h_out (directly into d_out)
  k_self128<<<blkNH, TB, 0, stream>>>(bufB, dinv, bufA);
  k_scat128<<<blkWaveE, TB, 0, stream>>>(ei, bufB, dinv, bufA);
  k_gemm_ln_relu<<<blkGemm, TB, 0, stream>>>(bufA, W3e, g3, b3, h_out, HID);

  // edge classifier
  k_edge_cls<<<blkWaveE, TB, 0, stream>>>(ei, h_out, Wce, logits);
}